// MultiHeadAggregator_55825984913801
// MI455X (gfx1250) — compile-verified
//
#include <hip/hip_runtime.h>
#include <hip/hip_bf16.h>
#include <math.h>

#define LNUM 12
#define BNUM 8
#define TNUM 2048
#define HNUM 768
#define CNUM 4

typedef __attribute__((ext_vector_type(2))) float v2f;
typedef __attribute__((ext_vector_type(8))) float v8f;

// ---------------------------------------------------------------------------
// Kernel 1: build compacted masked-position lists per batch.
// Mask 0 = HTE window [pos, pos+window] ; Mask 1 = descriptor tokens (or {0}).
// Serial per-batch scan (2048 ints) -> fully deterministic ordering.
// ---------------------------------------------------------------------------
__global__ void mha_masks_kernel(const int* __restrict__ ids,
                                 const int* __restrict__ hteTok,
                                 const int* __restrict__ dtok, int nd,
                                 const int* __restrict__ winPtr,
                                 int* __restrict__ cnt,
                                 int* __restrict__ posArr) {
    int b = blockIdx.x;
    if (threadIdx.x != 0) return;
    const int* row = ids + (size_t)b * TNUM;
    const int hte = *hteTok;
    const int window = *winPtr;
    int htePos = TNUM - 1;
    bool found = false;
    int dcount = 0;
    int* dlist = posArr + (size_t)(BNUM + b) * TNUM;   // mask 1 region
    for (int t = 0; t < TNUM; ++t) {
        int v = row[t];
        if (!found && v == hte) { htePos = t; found = true; }
        bool isd = false;
        for (int j = 0; j < nd; ++j) isd |= (v == dtok[j]);
        if (isd) dlist[dcount++] = t;
    }
    if (dcount == 0) { dlist[0] = 0; dcount = 1; }     // fallback: t == 0
    int pend = htePos + window;
    if (pend > TNUM - 1) pend = TNUM - 1;
    int wcount = pend - htePos + 1;
    int* wlist = posArr + (size_t)b * TNUM;            // mask 0 region
    for (int i = 0; i < wcount; ++i) wlist[i] = htePos + i;
    cnt[b] = wcount;
    cnt[BNUM + b] = dcount;
}

// ---------------------------------------------------------------------------
// Kernel 2: scores = X[pos] . Wt via V_WMMA_F32_16X16X4_F32, then softmax
// over the compacted positions per (layer, channel). One block per
// (mask, batch, layer); 8 waves each own 16-row score tiles.
// A fragment (16x4 f32, 2 VGPR): lanes 0-15 -> K={0,1}, lanes 16-31 -> K={2,3}
// B fragment (4x16 f32, 2 VGPR): N = lane&15, same K split.
// D fragment (16x16 f32, 8 VGPR): VGPR r -> M = r (+8 for upper lanes), N = lane&15.
// ---------------------------------------------------------------------------
__global__ __launch_bounds__(256)
void mha_scores_kernel(const float* __restrict__ layers,
                       const float* __restrict__ sw,
                       const int* __restrict__ cnt,
                       const int* __restrict__ posArr,
                       float* __restrict__ attnG) {
    __shared__ float sc[TNUM * CNUM];     // 32 KB score staging
    __shared__ float red[256];
    __shared__ float stat[2 * CNUM];      // [0..3]=max, [4..7]=sum

    const int bx   = blockIdx.x;
    const int l    = bx % LNUM;
    const int rem  = bx / LNUM;
    const int b    = rem % BNUM;
    const int mask = rem / BNUM;

    const int M = cnt[mask * BNUM + b];
    const int* plist = posArr + (size_t)(mask * BNUM + b) * TNUM;
    const float* xbase = layers + ((size_t)l * BNUM + b) * TNUM * HNUM;

    const int tid  = threadIdx.x;
    const int lane = tid & 31;
    const int wave = tid >> 5;
    const int row  = lane & 15;
    const int khi  = (lane < 16) ? 0 : 2;
    const int n    = lane & 15;
    const float nmsk = (n < CNUM) ? 1.0f : 0.0f;       // zero out unused B columns
    const float* wr = sw + (size_t)(n & (CNUM - 1)) * HNUM;  // always in-bounds

    const int numTiles = (M + 15) >> 4;
    for (int tile = wave; tile < numTiles; tile += 8) {
        int m  = tile * 16 + row;
        int mc = (m < M) ? m : (M - 1);                // clamp pad rows
        const float* rowp = xbase + (size_t)plist[mc] * HNUM;
        v8f acc0 = {};
        v8f acc1 = {};
        #pragma unroll 4
        for (int k0 = 0; k0 < HNUM; k0 += 8) {
            v2f a0, a1, b0, b1;
            a0.x = rowp[k0 + khi];         a0.y = rowp[k0 + khi + 1];
            a1.x = rowp[k0 + 4 + khi];     a1.y = rowp[k0 + 4 + khi + 1];
            b0.x = wr[k0 + khi] * nmsk;    b0.y = wr[k0 + khi + 1] * nmsk;
            b1.x = wr[k0 + 4 + khi] * nmsk; b1.y = wr[k0 + 4 + khi + 1] * nmsk;
            acc0 = __builtin_amdgcn_wmma_f32_16x16x4_f32(false, a0, false, b0,
                                                         (short)0, acc0, false, false);
            acc1 = __builtin_amdgcn_wmma_f32_16x16x4_f32(false, a1, false, b1,
                                                         (short)0, acc1, false, false);
        }
        v8f acc = acc0 + acc1;
        if (n < CNUM) {
            #pragma unroll
            for (int r = 0; r < 8; ++r) {
                int mr = tile * 16 + r + ((lane < 16) ? 0 : 8);
                if (mr < M) sc[mr * CNUM + n] = acc[r];
            }
        }
    }
    __syncthreads();

    // Softmax over compacted positions per channel (== reference masked softmax).
    for (int c = 0; c < CNUM; ++c) {
        float pm = -3.0e38f;
        for (int m = tid; m < M; m += 256) pm = fmaxf(pm, sc[m * CNUM + c]);
        red[tid] = pm; __syncthreads();
        for (int s = 128; s > 0; s >>= 1) {
            if (tid < s) red[tid] = fmaxf(red[tid], red[tid + s]);
            __syncthreads();
        }
        if (tid == 0) stat[c] = red[0];
        __syncthreads();
        float ps = 0.0f;
        for (int m = tid; m < M; m += 256) ps += __expf(sc[m * CNUM + c] - stat[c]);
        red[tid] = ps; __syncthreads();
        for (int s = 128; s > 0; s >>= 1) {
            if (tid < s) red[tid] += red[tid + s];
            __syncthreads();
        }
        if (tid == 0) stat[CNUM + c] = red[0];
        __syncthreads();
    }

    float* ag = attnG + ((size_t)(mask * BNUM + b) * LNUM + l) * TNUM * CNUM;
    for (int i = tid; i < M * CNUM; i += 256) {
        int c = i & (CNUM - 1);
        ag[i] = __expf(sc[i] - stat[c]) / stat[CNUM + c];
    }
}

// ---------------------------------------------------------------------------
// Kernel 3: pooled[c,h] = sum_l lw[l] * sum_m attn[l,m,c] * X[l,pos_m,h].
// One block per (mask, batch). Threads own h columns -> fully coalesced
// 3 KB row reads of the gathered positions; layer-weighted attention staged
// in LDS; next gathered row prefetched (global_prefetch_b8).
// ---------------------------------------------------------------------------
__global__ __launch_bounds__(256)
void mha_pool_kernel(const float* __restrict__ layers,
                     const float* __restrict__ layerLogits,
                     const int* __restrict__ cnt,
                     const int* __restrict__ posArr,
                     const float* __restrict__ attnG,
                     float* __restrict__ out) {
    __shared__ float lw[LNUM];
    __shared__ float aw[TNUM * CNUM];     // 32 KB attention staging
    __shared__ int   posL[TNUM];          // 8 KB positions

    const int bx   = blockIdx.x;
    const int b    = bx % BNUM;
    const int mask = bx / BNUM;
    const int tid  = threadIdx.x;

    if (tid == 0) {   // layer softmax (12 elements, redundant per block, trivial)
        float mx = layerLogits[0];
        for (int l = 1; l < LNUM; ++l) mx = fmaxf(mx, layerLogits[l]);
        float s = 0.0f;
        for (int l = 0; l < LNUM; ++l) s += __expf(layerLogits[l] - mx);
        for (int l = 0; l < LNUM; ++l) lw[l] = __expf(layerLogits[l] - mx) / s;
    }
    const int M = cnt[mask * BNUM + b];
    const int* plist = posArr + (size_t)(mask * BNUM + b) * TNUM;
    for (int i = tid; i < M; i += 256) posL[i] = plist[i];
    __syncthreads();

    float acc[3][CNUM];
    #pragma unroll
    for (int j = 0; j < 3; ++j)
        #pragma unroll
        for (int c = 0; c < CNUM; ++c) acc[j][c] = 0.0f;

    for (int l = 0; l < LNUM; ++l) {
        const float* ag = attnG + ((size_t)(mask * BNUM + b) * LNUM + l) * TNUM * CNUM;
        const float wl = lw[l];
        __syncthreads();
        for (int i = tid; i < M * CNUM; i += 256) aw[i] = ag[i] * wl;
        __syncthreads();
        const float* xb = layers + ((size_t)l * BNUM + b) * TNUM * HNUM;
        for (int m = 0; m < M; ++m) {
            const float* rowp = xb + (size_t)posL[m] * HNUM;
            if (m + 1 < M)
                __builtin_prefetch(xb + (size_t)posL[m + 1] * HNUM + tid, 0, 1);
            float a0 = aw[m * CNUM + 0], a1 = aw[m * CNUM + 1];
            float a2 = aw[m * CNUM + 2], a3 = aw[m * CNUM + 3];
            #pragma unroll
            for (int j = 0; j < 3; ++j) {
                float x = rowp[tid + j * 256];
                acc[j][0] += a0 * x; acc[j][1] += a1 * x;
                acc[j][2] += a2 * x; acc[j][3] += a3 * x;
            }
        }
    }

    float* ob = out + (size_t)b * (2 * CNUM * HNUM) + (size_t)mask * (CNUM * HNUM);
    #pragma unroll
    for (int j = 0; j < 3; ++j) {
        int h = tid + j * 256;
        #pragma unroll
        for (int c = 0; c < CNUM; ++c) ob[c * HNUM + h] = acc[j][c];
    }
}

// ---------------------------------------------------------------------------
extern "C" void kernel_launch(void* const* d_in, const int* in_sizes, int n_in,
                              void* d_out, int out_size, void* d_ws, size_t ws_size,
                              hipStream_t stream) {
    const float* hs   = (const float*)d_in[0];   // (13, B, T, H) fp32
    const int*   ids  = (const int*)d_in[1];     // (B, T)
    const float* ll   = (const float*)d_in[2];   // (12,)
    const float* sw   = (const float*)d_in[3];   // (C, H)
    const int*   hte  = (const int*)d_in[4];     // scalar
    const int*   dtok = (const int*)d_in[5];     // (3,)
    const int*   win  = (const int*)d_in[6];     // scalar
    const int nd = in_sizes[5];

    const float* layers = hs + (size_t)BNUM * TNUM * HNUM;   // skip hidden_states[0]

    // Workspace layout:
    //   [0, 64)                : counts  int[2][B]
    //   [256, 256+128K)        : posArr  int[2][B][T]
    //   [.., ..+6.0M)          : attnG   float[2][B][L][T][C]
    int*   cnt    = (int*)d_ws;
    int*   posArr = (int*)((char*)d_ws + 256);
    float* attnG  = (float*)((char*)d_ws + 256 + sizeof(int) * (size_t)2 * BNUM * TNUM);

    mha_masks_kernel<<<BNUM, 32, 0, stream>>>(ids, hte, dtok, nd, win, cnt, posArr);
    mha_scores_kernel<<<2 * BNUM * LNUM, 256, 0, stream>>>(layers, sw, cnt, posArr, attnG);
    mha_pool_kernel<<<2 * BNUM, 256, 0, stream>>>(layers, ll, cnt, posArr, attnG,
                                                  (float*)d_out);
}